// GraphMaxPool_11424613008099
// MI455X (gfx1250) — compile-verified
//
#include <hip/hip_runtime.h>
#include <hip/hip_bf16.h>

// Problem constants from the reference.
constexpr int B = 16;
constexpr int N = 4096;
constexpr int M = 1024;
constexpr int K = 32;    // == wave32 width: one index per lane
constexpr int F = 128;   // == 32 lanes * float4: one b128 load per lane per row

constexpr int WAVES_PER_BLOCK = 8;   // 256 threads = 8 waves (wave32)
constexpr int GROUP = 8;             // b128 loads in flight per wave

// float4 max: 4 independent v_max_num_f32 (VOPD dual-issue friendly).
__device__ __forceinline__ float4 fmax4(float4 a, float4 b) {
    float4 r;
    r.x = fmaxf(a.x, b.x);
    r.y = fmaxf(a.y, b.y);
    r.z = fmaxf(a.z, b.z);
    r.w = fmaxf(a.w, b.w);
    return r;
}

__global__ __launch_bounds__(WAVES_PER_BLOCK * 32)
void graph_max_pool_kernel(const float* __restrict__ in,
                           const int*   __restrict__ idx,
                           float*       __restrict__ out)
{
    const int lane = threadIdx.x & 31;
    const int task = blockIdx.x * WAVES_PER_BLOCK + (threadIdx.x >> 5); // b*M + m
    const int b = task >> 10;  // task / M (M = 1024)

    // Lane k holds batch_index[b][m][k] — one coalesced 128B load per wave.
    const int myidx = idx[(size_t)task * K + lane];

    const float* __restrict__ base = in + (size_t)b * ((size_t)N * F);
    const int foff = lane * 4;  // each lane owns 4 consecutive floats of F

    float4 acc;
    acc.x = acc.y = acc.z = acc.w = -__builtin_inff();

#pragma unroll
    for (int g = 0; g < K / GROUP; ++g) {
        // Prefetch the NEXT group one stage ahead (global_prefetch_b8: no VGPR
        // dest, no LOADcnt) while this group's loads drain.
        if (g + 1 < K / GROUP) {
#pragma unroll
            for (int j = 0; j < GROUP; ++j) {
                const int row =
                    __builtin_amdgcn_readlane(myidx, (g + 1) * GROUP + j);
                __builtin_prefetch((const void*)(base + (size_t)row * F + foff),
                                   0, 3);
            }
        }

        // Issue GROUP coalesced 512B row reads (global_load_b128; row base is
        // scalar via v_readlane_b32 with immediate lane, voffset = lane*16).
        float4 v[GROUP];
#pragma unroll
        for (int j = 0; j < GROUP; ++j) {
            const int row = __builtin_amdgcn_readlane(myidx, g * GROUP + j);
            v[j] = *(const float4*)(base + (size_t)row * F + foff);
        }

        // float4-granular tree reduction: depth 3 + acc, components adjacent.
        float4 m01 = fmax4(v[0], v[1]);
        float4 m23 = fmax4(v[2], v[3]);
        float4 m45 = fmax4(v[4], v[5]);
        float4 m67 = fmax4(v[6], v[7]);
        float4 m03 = fmax4(m01, m23);
        float4 m47 = fmax4(m45, m67);
        acc = fmax4(acc, fmax4(m03, m47));
    }

    // One coalesced 512B store per wave.
    *(float4*)(out + (size_t)task * F + foff) = acc;
}

extern "C" void kernel_launch(void* const* d_in, const int* in_sizes, int n_in,
                              void* d_out, int out_size, void* d_ws, size_t ws_size,
                              hipStream_t stream)
{
    (void)in_sizes; (void)n_in; (void)out_size; (void)d_ws; (void)ws_size;

    const float* in  = (const float*)d_in[0];   // [B, N, F] float32
    const int*   idx = (const int*)d_in[1];     // [B, M, K] integer indices
    float*       out = (float*)d_out;           // [B, M, F] float32

    const int total_waves = B * M;                                    // 16384
    const int blocks = total_waves / WAVES_PER_BLOCK;                 // 2048

    graph_max_pool_kernel<<<blocks, WAVES_PER_BLOCK * 32, 0, stream>>>(in, idx, out);
}